// HeteroGNN_26663156973732
// MI455X (gfx1250) — compile-verified
//
#include <hip/hip_runtime.h>

#define N_M 100000
#define N_D 30000
#define N_A 80000
#define DIN 64

typedef __attribute__((ext_vector_type(2))) float v2f;
typedef __attribute__((ext_vector_type(8))) float v8f;

// ---------------------------------------------------------------------------
// Edge scatter-add: agg[dst] += x[src] * ew   (d = 64, 16 threads per edge,
// each thread owns 4 consecutive features -> one b128 gather + 4 f32 atomics).
// unsafeAtomicAdd forces native global_atomic_add_f32 (no CAS expansion).
// ---------------------------------------------------------------------------
__global__ __launch_bounds__(256) void scatter_kernel(
    const float* __restrict__ x, const int* __restrict__ src,
    const int* __restrict__ dst, const float* __restrict__ ew,
    float* __restrict__ agg, int nedges)
{
    int tid = blockIdx.x * blockDim.x + threadIdx.x;
    int e = tid >> 4;
    if (e >= nedges) return;
    int f = (tid & 15) << 2;
    int s = src[e];
    int d = dst[e];
    float w = ew[e];
    const float4 v = *reinterpret_cast<const float4*>(x + (size_t)s * DIN + f);
    float* o = agg + (size_t)d * DIN + f;
    unsafeAtomicAdd(o + 0, v.x * w);
    unsafeAtomicAdd(o + 1, v.y * w);
    unsafeAtomicAdd(o + 2, v.z * w);
    unsafeAtomicAdd(o + 3, v.w * w);
}

// ---------------------------------------------------------------------------
// Fused node-wise linear:
//   out = agg0 @ Wrel0 [+ agg1 @ Wrel1] + x @ (Wroot0 [+ Wroot1]) + (b0 [+ b1])
// using V_WMMA_F32_16X16X4_F32 (full fp32). One wave32 per 16-row tile.
//   A fragment (16x4 f32): lane m=lane&15 is row; lanes 0-15 carry K=k4+{0,1},
//   lanes 16-31 carry K=k4+{2,3}  (ISA 7.12.2).  B mirrored by K.  C/D: VGPR r
//   -> row base+r (lanes 0-15) / base+r+8 (lanes 16-31), col = j*16 + (lane&15).
// Weights/bias staged in LDS once per block; root matrices pre-summed.
// ---------------------------------------------------------------------------
template<int DOUT, int NREL, bool RELU>
__global__ __launch_bounds__(256) void linear_kernel(
    const float* __restrict__ agg0, const float* __restrict__ agg1,
    const float* __restrict__ xroot,
    const float* __restrict__ Wrel0, const float* __restrict__ Wrel1,
    const float* __restrict__ Wroot0, const float* __restrict__ Wroot1,
    const float* __restrict__ bias0, const float* __restrict__ bias1,
    float* __restrict__ out, int n)
{
    constexpr int NSRC = NREL + 1;
    constexpr int WSZ  = DIN * DOUT;
    __shared__ float sW[NSRC * WSZ];
    __shared__ float sB[DOUT];

    for (int i = threadIdx.x; i < WSZ; i += blockDim.x) {
        sW[i] = Wrel0[i];
        if (NREL == 2) sW[WSZ + i] = Wrel1[i];
        float r = Wroot0[i];
        if (NREL == 2) r += Wroot1[i];
        sW[NREL * WSZ + i] = r;
    }
    for (int i = threadIdx.x; i < DOUT; i += blockDim.x) {
        float b = bias0[i];
        if (NREL == 2) b += bias1[i];
        sB[i] = b;
    }
    __syncthreads();

    const int lane = threadIdx.x & 31;
    const int wave = threadIdx.x >> 5;
    const int m    = lane & 15;
    const int koff = (lane >> 4) * 2;
    const int ntiles = n >> 4;            // all N are multiples of 16

    const float* srcs[NSRC];
    srcs[0] = agg0;
    if (NREL == 2) srcs[1] = agg1;
    srcs[NSRC - 1] = xroot;

    for (int t = blockIdx.x * 8 + wave; t < ntiles; t += gridDim.x * 8) {
        const int row = (t << 4) + m;
        // Cache all A fragments for this 16-row tile in registers.
        v2f afrag[NSRC][16];
        #pragma unroll
        for (int s = 0; s < NSRC; ++s) {
            const float* rp = srcs[s] + (size_t)row * DIN + koff;
            #pragma unroll
            for (int k = 0; k < 16; ++k)
                afrag[s][k] = *reinterpret_cast<const v2f*>(rp + k * 4);
        }
        #pragma unroll
        for (int j = 0; j < DOUT / 16; ++j) {
            const int col = j * 16 + m;
            const float bv = sB[col];
            v8f acc = {bv, bv, bv, bv, bv, bv, bv, bv};
            #pragma unroll
            for (int s = 0; s < NSRC; ++s) {
                const float* wp = sW + s * WSZ + col;
                #pragma unroll
                for (int k = 0; k < 16; ++k) {
                    const int kr = k * 4 + koff;
                    v2f b;
                    b.x = wp[kr * DOUT];
                    b.y = wp[(kr + 1) * DOUT];
                    acc = __builtin_amdgcn_wmma_f32_16x16x4_f32(
                        false, afrag[s][k], false, b, (short)0, acc, false, false);
                }
            }
            #pragma unroll
            for (int r = 0; r < 8; ++r) {
                float v = acc[r];
                if (RELU) v = fmaxf(v, 0.0f);
                const int orow = (t << 4) + r + ((lane >> 4) << 3);
                out[(size_t)orow * DOUT + j * 16 + m] = v;
            }
        }
    }
}

// ---------------------------------------------------------------------------
extern "C" void kernel_launch(void* const* d_in, const int* in_sizes, int n_in,
                              void* d_out, int out_size, void* d_ws, size_t ws_size,
                              hipStream_t stream) {
    const float* x_m    = (const float*)d_in[0];
    const float* x_d    = (const float*)d_in[1];
    const float* x_a    = (const float*)d_in[2];
    const int*   src_md = (const int*)d_in[3];
    const int*   dst_md = (const int*)d_in[4];
    const float* ew_md  = (const float*)d_in[5];
    const int*   src_dm = (const int*)d_in[6];
    const int*   dst_dm = (const int*)d_in[7];
    const float* ew_dm  = (const float*)d_in[8];
    const int*   src_ma = (const int*)d_in[9];
    const int*   dst_ma = (const int*)d_in[10];
    const float* ew_ma  = (const float*)d_in[11];
    const int*   src_am = (const int*)d_in[12];
    const int*   dst_am = (const int*)d_in[13];
    const float* ew_am  = (const float*)d_in[14];
    const float* W1_rel  = (const float*)d_in[15];
    const float* b1      = (const float*)d_in[16];
    const float* W1_root = (const float*)d_in[17];
    const float* W2_rel  = (const float*)d_in[18];
    const float* b2      = (const float*)d_in[19];
    const float* W2_root = (const float*)d_in[20];

    const int E_md = in_sizes[3];
    const int E_dm = in_sizes[6];
    const int E_ma = in_sizes[9];
    const int E_am = in_sizes[12];

    // Workspace layout (floats)
    float* ws     = (float*)d_ws;
    float* agg_md = ws;                                // [N_D,64]
    float* agg_dm = agg_md + (size_t)N_D * DIN;        // [N_M,64]
    float* agg_ma = agg_dm + (size_t)N_M * DIN;        // [N_A,64]
    float* agg_am = agg_ma + (size_t)N_A * DIN;        // [N_M,64]
    float* h_m    = agg_am + (size_t)N_M * DIN;        // [N_M,64]
    float* h_d    = h_m    + (size_t)N_M * DIN;        // [N_D,64]
    float* h_a    = h_d    + (size_t)N_D * DIN;        // [N_A,64]
    const size_t aggBytes = (size_t)(N_D + N_M + N_A + N_M) * DIN * sizeof(float);

    float* out = (float*)d_out;
    float* o_m = out;
    float* o_d = out + (size_t)N_M * 32;
    float* o_a = out + (size_t)(N_M + N_D) * 32;

    auto sc = [&](const float* x, const int* s, const int* d, const float* w,
                  float* agg, int ne) {
        int blocks = (ne * 16 + 255) / 256;
        scatter_kernel<<<blocks, 256, 0, stream>>>(x, s, d, w, agg, ne);
    };
    auto grid1 = [](int n) {
        int b = (n / 16 + 7) / 8;
        return b > 1024 ? 1024 : b;
    };

    const int W1 = 64 * 64, B1 = 64, W2 = 64 * 32, B2 = 32;

    // ---------------- Layer 1 ----------------
    hipMemsetAsync(agg_md, 0, aggBytes, stream);
    sc(x_m, src_md, dst_md, ew_md, agg_md, E_md);
    sc(x_d, src_dm, dst_dm, ew_dm, agg_dm, E_dm);
    sc(x_m, src_ma, dst_ma, ew_ma, agg_ma, E_ma);
    sc(x_a, src_am, dst_am, ew_am, agg_am, E_am);

    linear_kernel<64, 1, true><<<grid1(N_D), 256, 0, stream>>>(
        agg_md, nullptr, x_d,
        W1_rel + 0 * W1, nullptr, W1_root + 0 * W1, nullptr,
        b1 + 0 * B1, nullptr, h_d, N_D);
    linear_kernel<64, 2, true><<<grid1(N_M), 256, 0, stream>>>(
        agg_dm, agg_am, x_m,
        W1_rel + 1 * W1, W1_rel + 3 * W1, W1_root + 1 * W1, W1_root + 3 * W1,
        b1 + 1 * B1, b1 + 3 * B1, h_m, N_M);
    linear_kernel<64, 1, true><<<grid1(N_A), 256, 0, stream>>>(
        agg_ma, nullptr, x_a,
        W1_rel + 2 * W1, nullptr, W1_root + 2 * W1, nullptr,
        b1 + 2 * B1, nullptr, h_a, N_A);

    // ---------------- Layer 2 ----------------
    hipMemsetAsync(agg_md, 0, aggBytes, stream);
    sc(h_m, src_md, dst_md, ew_md, agg_md, E_md);
    sc(h_d, src_dm, dst_dm, ew_dm, agg_dm, E_dm);
    sc(h_m, src_ma, dst_ma, ew_ma, agg_ma, E_ma);
    sc(h_a, src_am, dst_am, ew_am, agg_am, E_am);

    linear_kernel<32, 1, false><<<grid1(N_D), 256, 0, stream>>>(
        agg_md, nullptr, h_d,
        W2_rel + 0 * W2, nullptr, W2_root + 0 * W2, nullptr,
        b2 + 0 * B2, nullptr, o_d, N_D);
    linear_kernel<32, 2, false><<<grid1(N_M), 256, 0, stream>>>(
        agg_dm, agg_am, h_m,
        W2_rel + 1 * W2, W2_rel + 3 * W2, W2_root + 1 * W2, W2_root + 3 * W2,
        b2 + 1 * B2, b2 + 3 * B2, o_m, N_M);
    linear_kernel<32, 1, false><<<grid1(N_A), 256, 0, stream>>>(
        agg_ma, nullptr, h_a,
        W2_rel + 2 * W2, nullptr, W2_root + 2 * W2, nullptr,
        b2 + 2 * B2, nullptr, o_a, N_A);
}